// PartialLabelwiseAttention_61031485276563
// MI455X (gfx1250) — compile-verified
//
#include <hip/hip_runtime.h>

typedef __attribute__((ext_vector_type(2))) float v2f;
typedef __attribute__((ext_vector_type(8))) float v8f;

#define B_ 16
#define T_ 1024
#define H_ 256
#define L_ 2048

// -------------------------------------------------------------------------
// Pass 1: S = Q @ X^T (per batch), softmax over T, write attention (B,L,T)
// Block: 256 threads = 8 waves. Block tile: 32 L-rows x full T (1024).
// Each wave owns 128 T-columns; 2 M-tiles x 8 N-tiles = 16 WMMA C tiles.
// Each X (B-matrix) fragment load feeds 2 WMMAs -> halves L2 traffic.
// -------------------------------------------------------------------------
__global__ __launch_bounds__(256) void attn_scores_softmax_kernel(
    const float* __restrict__ X,           // (B,T,H)
    const long long* __restrict__ labels,  // (B,L) int64
    const float* __restrict__ table,       // (C+1,H)
    float* __restrict__ attn)              // (B,L,T) output slice
{
  __shared__ float Qs[32 * H_];   // 32KB: gathered query rows
  __shared__ float red[8][32];    // cross-wave row reductions (8 waves x 32 rows)

  const int tid  = threadIdx.x;
  const int wave = tid >> 5;
  const int lane = tid & 31;
  const int half = lane >> 4;     // lane-half: selects K pair {0,1}/{2,3}, M half
  const int ln   = lane & 15;
  const int b    = blockIdx.y;
  const int l0   = blockIdx.x << 5;     // 32 L-rows per block

  // ---- stage Q tile: gather 32 rows of the table into LDS (coalesced) ----
  {
    const int r  = tid >> 3;              // 0..31 (L row), 8 threads per row
    const int c0 = (tid & 7) << 5;        // 32 floats per thread
    const long long lab = labels[(size_t)b * L_ + l0 + r];
    const float4* src = (const float4*)(table + (size_t)lab * H_ + c0);
    float4* dst = (float4*)(&Qs[r * H_ + c0]);
    #pragma unroll
    for (int i = 0; i < 8; ++i) dst[i] = src[i];
  }
  __syncthreads();

  const float* Xb = X + (size_t)b * T_ * H_;

  v8f acc[2][8];
  #pragma unroll
  for (int mi = 0; mi < 2; ++mi)
    #pragma unroll
    for (int n = 0; n < 8; ++n) acc[mi][n] = (v8f)0.0f;

  // ---- K loop over H in steps of 4 (V_WMMA_F32_16X16X4_F32, exact f32) ----
  #pragma unroll 1
  for (int k = 0; k < H_; k += 4) {
    const int kb = k + (half << 1);                 // K pair: {0,1} or {2,3}
    // A fragments: rows ln and 16+ln of Q, K = kb, kb+1 (LDS, 8B aligned)
    v2f a0 = *(const v2f*)&Qs[ln * H_ + kb];
    v2f a1 = *(const v2f*)&Qs[(16 + ln) * H_ + kb];
    #pragma unroll
    for (int n = 0; n < 8; ++n) {
      const int t = (wave << 7) + (n << 4) + ln;    // N column = T index
      const float* xp = Xb + (size_t)t * H_ + kb;   // B[k][t] = X[t][k]
      v2f bb; bb.x = xp[0]; bb.y = xp[1];
      acc[0][n] = __builtin_amdgcn_wmma_f32_16x16x4_f32(
          false, a0, false, bb, (short)0, acc[0][n], false, false);
      acc[1][n] = __builtin_amdgcn_wmma_f32_16x16x4_f32(
          false, a1, false, bb, (short)0, acc[1][n], false, false);
    }
  }

  // ---- softmax over T: row m = mi*16 + 8*half + v lives in acc[mi][n][v] ----
  // 1) row max: per-lane over n, across 16-lane half (shfl), then waves (LDS)
  float gmax[2][8];
  #pragma unroll
  for (int mi = 0; mi < 2; ++mi) {
    #pragma unroll
    for (int v = 0; v < 8; ++v) {
      float m = acc[mi][0][v];
      #pragma unroll
      for (int n = 1; n < 8; ++n) m = fmaxf(m, acc[mi][n][v]);
      #pragma unroll
      for (int off = 1; off < 16; off <<= 1)
        m = fmaxf(m, __shfl_xor(m, off, 32));
      gmax[mi][v] = m;
    }
  }
  if (ln == 0) {
    #pragma unroll
    for (int mi = 0; mi < 2; ++mi)
      #pragma unroll
      for (int v = 0; v < 8; ++v)
        red[wave][(mi << 4) + (half << 3) + v] = gmax[mi][v];
  }
  __syncthreads();
  #pragma unroll
  for (int mi = 0; mi < 2; ++mi) {
    #pragma unroll
    for (int v = 0; v < 8; ++v) {
      float m = red[0][(mi << 4) + (half << 3) + v];
      #pragma unroll
      for (int w = 1; w < 8; ++w)
        m = fmaxf(m, red[w][(mi << 4) + (half << 3) + v]);
      gmax[mi][v] = m;
    }
  }
  __syncthreads();

  // 2) exp + row sum
  float gsum[2][8];
  #pragma unroll
  for (int mi = 0; mi < 2; ++mi) {
    #pragma unroll
    for (int v = 0; v < 8; ++v) {
      float s = 0.0f;
      #pragma unroll
      for (int n = 0; n < 8; ++n) {
        float e = __expf(acc[mi][n][v] - gmax[mi][v]);
        acc[mi][n][v] = e;
        s += e;
      }
      #pragma unroll
      for (int off = 1; off < 16; off <<= 1)
        s += __shfl_xor(s, off, 32);
      gsum[mi][v] = s;
    }
  }
  if (ln == 0) {
    #pragma unroll
    for (int mi = 0; mi < 2; ++mi)
      #pragma unroll
      for (int v = 0; v < 8; ++v)
        red[wave][(mi << 4) + (half << 3) + v] = gsum[mi][v];
  }
  __syncthreads();
  #pragma unroll
  for (int mi = 0; mi < 2; ++mi) {
    #pragma unroll
    for (int v = 0; v < 8; ++v) {
      float s = 0.0f;
      #pragma unroll
      for (int w = 0; w < 8; ++w)
        s += red[w][(mi << 4) + (half << 3) + v];
      gsum[mi][v] = s;
    }
  }

  // 3) normalize + store attention (64B contiguous per half-wave per store)
  float* ap = attn + ((size_t)b * L_ + l0) * T_;
  #pragma unroll
  for (int mi = 0; mi < 2; ++mi) {
    #pragma unroll
    for (int v = 0; v < 8; ++v) {
      const float inv = 1.0f / gsum[mi][v];
      const int row = (mi << 4) + (half << 3) + v;
      #pragma unroll
      for (int n = 0; n < 8; ++n) {
        const int t = (wave << 7) + (n << 4) + ln;
        ap[(size_t)row * T_ + t] = acc[mi][n][v] * inv;
      }
    }
  }
}

// -------------------------------------------------------------------------
// Pass 2: logits = P @ X (per batch). Block tile: 32 L-rows x full H (256).
// 8 waves x 32 H-cols; 2 M-tiles x 2 N-tiles = 4 WMMA C tiles per wave.
// P staged in 32KB LDS chunks (coalesced); attention is L2-resident (192MB).
// -------------------------------------------------------------------------
__global__ __launch_bounds__(256) void attn_output_kernel(
    const float* __restrict__ X,     // (B,T,H)
    const float* __restrict__ attn,  // (B,L,T)
    float* __restrict__ out)         // (B,L,H) logits
{
  __shared__ float Ps[32 * 256];     // 32KB: 32 rows x 256 t-chunk

  const int tid  = threadIdx.x;
  const int wave = tid >> 5;
  const int lane = tid & 31;
  const int half = lane >> 4;
  const int ln   = lane & 15;
  const int b    = blockIdx.y;
  const int l0   = blockIdx.x << 5;

  const float* Xb = X + (size_t)b * T_ * H_;
  const float* Pp = attn + ((size_t)b * L_ + l0) * T_;
  const int h0 = wave << 5;          // 32 H columns per wave

  v8f acc[2][2];
  #pragma unroll
  for (int mi = 0; mi < 2; ++mi) {
    acc[mi][0] = (v8f)0.0f; acc[mi][1] = (v8f)0.0f;
  }

  #pragma unroll 1
  for (int k0 = 0; k0 < T_; k0 += 256) {
    // stage 32x256 chunk of P (coalesced float4 loads)
    {
      const int r  = tid >> 3;              // 0..31
      const int c0 = (tid & 7) << 5;        // 32 floats per thread
      const float4* src = (const float4*)(Pp + (size_t)r * T_ + k0 + c0);
      float4* dst = (float4*)&Ps[r * 256 + c0];
      #pragma unroll
      for (int i = 0; i < 8; ++i) dst[i] = src[i];
    }
    __syncthreads();

    #pragma unroll 1
    for (int k = 0; k < 256; k += 4) {
      const int kb = k + (half << 1);
      v2f a0 = *(const v2f*)&Ps[ln * 256 + kb];          // P rows 0..15
      v2f a1 = *(const v2f*)&Ps[(16 + ln) * 256 + kb];   // P rows 16..31
      const int kg = k0 + kb;
      #pragma unroll
      for (int n = 0; n < 2; ++n) {
        const int h = h0 + (n << 4) + ln;                // N column = H index
        v2f bb;                                          // B[k][h] = X[k][h]
        bb.x = Xb[(size_t)kg * H_ + h];
        bb.y = Xb[(size_t)(kg + 1) * H_ + h];
        acc[0][n] = __builtin_amdgcn_wmma_f32_16x16x4_f32(
            false, a0, false, bb, (short)0, acc[0][n], false, false);
        acc[1][n] = __builtin_amdgcn_wmma_f32_16x16x4_f32(
            false, a1, false, bb, (short)0, acc[1][n], false, false);
      }
    }
    __syncthreads();
  }

  // store logits: row = mi*16 + 8*half + v, col = h0 + n*16 + ln
  float* op = out + ((size_t)b * L_ + l0) * H_;
  #pragma unroll
  for (int mi = 0; mi < 2; ++mi) {
    #pragma unroll
    for (int v = 0; v < 8; ++v) {
      const int row = (mi << 4) + (half << 3) + v;
      #pragma unroll
      for (int n = 0; n < 2; ++n) {
        const int h = h0 + (n << 4) + ln;
        op[(size_t)row * H_ + h] = acc[mi][n][v];
      }
    }
  }
}

extern "C" void kernel_launch(void* const* d_in, const int* in_sizes, int n_in,
                              void* d_out, int out_size, void* d_ws, size_t ws_size,
                              hipStream_t stream) {
  const float*     X      = (const float*)d_in[0];     // inputs (B,T,H) f32
  const long long* labels = (const long long*)d_in[1]; // labels (B,L) i64
  const float*     table  = (const float*)d_in[2];     // attention_table f32

  float* out    = (float*)d_out;
  float* logits = out;                                  // first: (B,L,H)
  float* attn   = out + (size_t)B_ * L_ * H_;           // second: (B,L,T)

  dim3 grid(L_ / 32, B_);
  attn_scores_softmax_kernel<<<grid, 256, 0, stream>>>(X, labels, table, attn);
  attn_output_kernel<<<grid, 256, 0, stream>>>(X, attn, logits);
}